// VAE_59777354826409
// MI455X (gfx1250) — compile-verified
//
#include <hip/hip_runtime.h>
#include <math.h>

// ---------------- Problem constants ----------------
#define BB   32
#define SP   64
#define SH   64
#define EE   512
#define HH   1024
#define H4   4096
#define VV   32000
#define SD   63            // decoder sequence length (SH-1)
#define MDEC (BB*SD)       // 2016

// ---------------- WMMA vector types ----------------
typedef __attribute__((ext_vector_type(2))) float v2f;
typedef __attribute__((ext_vector_type(4))) float v4f;
typedef __attribute__((ext_vector_type(8))) float v8f;

// ---------------- GEMM tiling ----------------
#define BM 32
#define BN 128
#define KC 32
#define LAS 36           // padded row stride (mult of 4 -> b128-aligned, odd/4 -> bank-clean)
#define BNUM 4           // 4 x float4 of W per thread per chunk (1 float4 of A)

// C[m,n] = sum_k A[m,k] * W[n,k]  (+ bias1[n] + bias2[n]) (+ Cadd[m*ldadd + n])
// If REMAP: output row m is stored at row ((m/63)*64 + (m%63) + 1) of C.
// Block tile 32x128: 8 waves, wave w owns N-strip w*16, both M tiles (2 accumulators).
// Both A and W staged ROW-MAJOR over the K-chunk (WMMA B-frag reads W rows directly).
template <bool HAS_B1, bool HAS_B2, bool HAS_ADD, bool REMAP>
__launch_bounds__(256)
__global__ void k_gemm_xwT(const float* __restrict__ A, int lda,
                           const float* __restrict__ W,
                           const float* __restrict__ bias1,
                           const float* __restrict__ bias2,
                           const float* __restrict__ Cadd, int ldadd,
                           float* __restrict__ C, int ldc,
                           int K)
{
    __shared__ float As[BM * LAS];
    __shared__ float Bs[BN * LAS];

    const int tid  = threadIdx.x;
    const int m0   = blockIdx.y * BM;
    const int n0   = blockIdx.x * BN;
    const int w    = tid >> 5;          // wave id 0..7 (wave32)
    const int lane = tid & 31;
    const int hi   = lane >> 4;         // lane half -> k pair select
    const int lid  = lane & 15;
    const int nt   = w << 4;            // wave's N strip within block

    // vectorized staging coords
    const int ar  = tid >> 3;           // A row 0..31
    const int ac4 = (tid & 7) << 2;     // A k-offset (float4)

    // register staging for global->LDS double buffering
    v4f ra, rb[BNUM];

    auto loadG = [&](int k0) {
        ra = *(const v4f*)(A + (size_t)(m0 + ar) * lda + (k0 + ac4));
#pragma unroll
        for (int i = 0; i < BNUM; ++i) {
            int fidx = tid + i * 256;
            int n = fidx >> 3, c4 = (fidx & 7) << 2;
            rb[i] = *(const v4f*)(W + (size_t)(n0 + n) * K + (k0 + c4));
        }
    };
    auto storeL = [&]() {
        *(v4f*)(As + ar * LAS + ac4) = ra;
#pragma unroll
        for (int i = 0; i < BNUM; ++i) {
            int fidx = tid + i * 256;
            int n = fidx >> 3, c4 = (fidx & 7) << 2;
            *(v4f*)(Bs + n * LAS + c4) = rb[i];
        }
    };

    v8f acc0 = {0.f, 0.f, 0.f, 0.f, 0.f, 0.f, 0.f, 0.f};
    v8f acc1 = {0.f, 0.f, 0.f, 0.f, 0.f, 0.f, 0.f, 0.f};

    loadG(0);
    for (int k0 = 0; k0 < K; k0 += KC) {
        storeL();
        __syncthreads();
        if (k0 + KC < K) loadG(k0 + KC);   // prefetch next chunk (overlaps WMMAs)

        // Preload ALL fragments for this chunk (b64 LDS reads), then chain WMMAs.
        v2f af0[KC / 4], af1[KC / 4], bf[KC / 4];
#pragma unroll
        for (int s = 0; s < KC / 4; ++s) {
            const int ka = (s << 2) + (hi << 1);
            af0[s] = *(const v2f*)(As + lid * LAS + ka);
            af1[s] = *(const v2f*)(As + (16 + lid) * LAS + ka);
            bf[s]  = *(const v2f*)(Bs + (nt + lid) * LAS + ka);
        }
#pragma unroll
        for (int s = 0; s < KC / 4; ++s) {
            acc0 = __builtin_amdgcn_wmma_f32_16x16x4_f32(
                false, af0[s], false, bf[s], (short)0, acc0, false, false);
            acc1 = __builtin_amdgcn_wmma_f32_16x16x4_f32(
                false, af1[s], false, bf[s], (short)0, acc1, false, false);
        }
        __syncthreads();
    }

    // Epilogue. C/D layout: VGPR r -> M = r + 8*hi ; N = lid
    const int n = n0 + nt + lid;
    float bsum = 0.f;
    if (HAS_B1) bsum += bias1[n];
    if (HAS_B2) bsum += bias2[n];
#pragma unroll
    for (int r = 0; r < 8; ++r) {
#pragma unroll
        for (int half = 0; half < 2; ++half) {
            int m = m0 + (half << 4) + (hi << 3) + r;
            float v = (half ? acc1[r] : acc0[r]) + bsum;
            if (HAS_ADD) v += Cadd[(size_t)m * ldadd + n];
            size_t orow;
            if (REMAP) {
                int bb = m / SD, ss = m % SD;
                orow = (size_t)(bb * SH + ss + 1);
            } else {
                orow = (size_t)m;
            }
            C[orow * (size_t)ldc + n] = v;
        }
    }
}

// ---------------- Elementwise kernels ----------------
__global__ void k_zero(float* __restrict__ p, int n) {
    int i = blockIdx.x * blockDim.x + threadIdx.x;
    if (i < n) p[i] = 0.f;
}

// zero raw_outputs[:,0,:]
__global__ void k_zero_row0(float* __restrict__ raw) {
    int b = blockIdx.x;                   // 0..31
    float* row = raw + (size_t)(b * SH) * VV;
    for (int j = threadIdx.x; j < VV; j += blockDim.x) row[j] = 0.f;
}

// Embedding gather: dst[b*S_use+s] = embed_W[tok[b*S_in+s]]
__global__ void k_gather(const float* __restrict__ embW,
                         const int* __restrict__ tok,
                         int S_in, int S_use,
                         float* __restrict__ dst)
{
    int m = blockIdx.x;
    int b = m / S_use, s = m % S_use;
    int t = tok[b * S_in + s];
    const v4f* src = (const v4f*)(embW + (size_t)t * EE);
    v4f* d = (v4f*)(dst + (size_t)m * EE);
    if (threadIdx.x < EE / 4) d[threadIdx.x] = src[threadIdx.x];
}

__device__ __forceinline__ float sigmoidf_(float x) { return 1.f / (1.f + __expf(-x)); }

// One encoder LSTM step: pre (32 x 4096) already has x_pre(t) + h@Whh^T
__global__ void k_lstm_step(const float* __restrict__ pre,
                            float* __restrict__ h, float* __restrict__ c)
{
    int idx = blockIdx.x * blockDim.x + threadIdx.x;   // 32768
    int b = idx >> 10, j = idx & (HH - 1);
    const float* p = pre + (size_t)b * H4;
    float gi = p[j], gf = p[HH + j], gg = p[2 * HH + j], go = p[3 * HH + j];
    float c2 = sigmoidf_(gf) * c[idx] + sigmoidf_(gi) * tanhf(gg);
    float h2 = sigmoidf_(go) * tanhf(c2);
    c[idx] = c2;
    h[idx] = h2;
}

// ---------------- threefry2x32 + erfinv (jax.random.normal(key(42))) ----
__device__ __forceinline__ unsigned rotl32_(unsigned x, int r) {
    return (x << r) | (x >> (32 - r));
}
__device__ void threefry2x32_(unsigned k0, unsigned k1, unsigned c0, unsigned c1,
                              unsigned& o0, unsigned& o1)
{
    unsigned ks2 = k0 ^ k1 ^ 0x1BD11BDAu;
    unsigned x0 = c0 + k0, x1 = c1 + k1;
#define TF_R(r) { x0 += x1; x1 = rotl32_(x1, r); x1 ^= x0; }
    TF_R(13) TF_R(15) TF_R(26) TF_R(6)
    x0 += k1;  x1 += ks2 + 1u;
    TF_R(17) TF_R(29) TF_R(16) TF_R(24)
    x0 += ks2; x1 += k0 + 2u;
    TF_R(13) TF_R(15) TF_R(26) TF_R(6)
    x0 += k0;  x1 += k1 + 3u;
    TF_R(17) TF_R(29) TF_R(16) TF_R(24)
    x0 += k1;  x1 += ks2 + 4u;
    TF_R(13) TF_R(15) TF_R(26) TF_R(6)
    x0 += ks2; x1 += k0 + 5u;
#undef TF_R
    o0 = x0; o1 = x1;
}
__device__ float erfinvf_(float x) {
    float w = -__logf((1.0f - x) * (1.0f + x));
    float p;
    if (w < 5.0f) {
        w -= 2.5f;
        p = 2.81022636e-08f;
        p = fmaf(p, w, 3.43273939e-07f);  p = fmaf(p, w, -3.5233877e-06f);
        p = fmaf(p, w, -4.39150654e-06f); p = fmaf(p, w, 0.00021858087f);
        p = fmaf(p, w, -0.00125372503f);  p = fmaf(p, w, -0.00417768164f);
        p = fmaf(p, w, 0.246640727f);     p = fmaf(p, w, 1.50140941f);
    } else {
        w = sqrtf(w) - 3.0f;
        p = -0.000200214257f;
        p = fmaf(p, w, 0.000100950558f);  p = fmaf(p, w, 0.00134934322f);
        p = fmaf(p, w, -0.00367342844f);  p = fmaf(p, w, 0.00573950773f);
        p = fmaf(p, w, -0.0076224613f);   p = fmaf(p, w, 0.00943887047f);
        p = fmaf(p, w, 1.00167406f);      p = fmaf(p, w, 2.83297682f);
    }
    return p * x;
}

// z0 = mu + eps * exp(0.5*lv), eps = jax.random.normal(key(42), (32,1024))
__global__ void k_sample_z(const float* __restrict__ mu, const float* __restrict__ lv,
                           float* __restrict__ z)
{
    int idx = blockIdx.x * blockDim.x + threadIdx.x;   // 32768
    const int half = (BB * HH) / 2;
    unsigned o0, o1, bits;
    if (idx < half) {
        threefry2x32_(0u, 42u, (unsigned)idx, (unsigned)(idx + half), o0, o1);
        bits = o0;
    } else {
        threefry2x32_(0u, 42u, (unsigned)(idx - half), (unsigned)idx, o0, o1);
        bits = o1;
    }
    unsigned fb = (bits >> 9) | 0x3f800000u;
    float u01;
    __builtin_memcpy(&u01, &fb, 4);
    float u = fmaf(u01 - 1.0f, 2.0f, -1.0f);           // uniform in [-1, 1)
    u = fmaxf(u, -0.9999999f);
    float eps = 1.41421356237f * erfinvf_(u);
    z[idx] = mu[idx] + eps * __expf(0.5f * lv[idx]);
}

// Decoder gates: pre1 (2016 x 4096) has emb@Wih^T + bih + bhh; add zW[b], c = z0[b]
__global__ void k_dec_gates(const float* __restrict__ pre1, const float* __restrict__ zW,
                            const float* __restrict__ z0, float* __restrict__ hdec)
{
    int idx = blockIdx.x * blockDim.x + threadIdx.x;   // 2016*1024
    int m = idx >> 10, j = idx & (HH - 1);
    int b = m / SD;
    const float* p = pre1 + (size_t)m * H4;
    const float* zw = zW + (size_t)b * H4;
    float gi = p[j] + zw[j];
    float gf = p[HH + j] + zw[HH + j];
    float gg = p[2 * HH + j] + zw[2 * HH + j];
    float go = p[3 * HH + j] + zw[3 * HH + j];
    float c2 = sigmoidf_(gf) * z0[b * HH + j] + sigmoidf_(gi) * tanhf(gg);
    hdec[idx] = sigmoidf_(go) * tanhf(c2);
}

// Argmax over V for each of 2016 logit rows -> outputted_words[b, s+1]
__global__ void k_argmax(const float* __restrict__ raw, float* __restrict__ ow)
{
    __shared__ float sv[256];
    __shared__ int   si[256];
    int m = blockIdx.x;
    int b = m / SD, s = m % SD;
    const float* row = raw + (size_t)(b * SH + s + 1) * VV;
    float best = -3.4e38f; int bi = 0;
    for (int j = threadIdx.x; j < VV; j += 256) {
        float v = row[j];
        if (v > best) { best = v; bi = j; }
    }
    sv[threadIdx.x] = best; si[threadIdx.x] = bi;
    __syncthreads();
    for (int o = 128; o > 0; o >>= 1) {
        if ((int)threadIdx.x < o) {
            float v2 = sv[threadIdx.x + o]; int i2 = si[threadIdx.x + o];
            if (v2 > sv[threadIdx.x] || (v2 == sv[threadIdx.x] && i2 < si[threadIdx.x])) {
                sv[threadIdx.x] = v2; si[threadIdx.x] = i2;
            }
        }
        __syncthreads();
    }
    if (threadIdx.x == 0) ow[b * SH + s + 1] = (float)si[0];
}

__global__ void k_ow0(const int* __restrict__ hyp, float* __restrict__ ow) {
    int b = threadIdx.x;
    if (b < BB) ow[b * SH] = (float)hyp[b * SH];
}

// ---------------- Host side ----------------
extern "C" void kernel_launch(void* const* d_in, const int* in_sizes, int n_in,
                              void* d_out, int out_size, void* d_ws, size_t ws_size,
                              hipStream_t stream)
{
    const int*   premises   = (const int*)d_in[0];
    const int*   hypothesis = (const int*)d_in[1];
    // d_in[2] = labels (unused)
    const float* embed_W  = (const float*)d_in[3];
    const float* enc_Wih  = (const float*)d_in[4];
    const float* enc_Whh  = (const float*)d_in[5];
    const float* enc_bih  = (const float*)d_in[6];
    const float* enc_bhh  = (const float*)d_in[7];
    const float* mu_W     = (const float*)d_in[8];
    const float* mu_b     = (const float*)d_in[9];
    const float* lv_W     = (const float*)d_in[10];
    const float* lv_b     = (const float*)d_in[11];
    const float* dec_Wih  = (const float*)d_in[12];
    const float* dec_Whh  = (const float*)d_in[13];
    const float* dec_bih  = (const float*)d_in[14];
    const float* dec_bhh  = (const float*)d_in[15];
    const float* out_W    = (const float*)d_in[16];
    const float* out_b    = (const float*)d_in[17];

    float* out = (float*)d_out;
    // d_out layout: outputted_words | raw_outputs | mu0 | log_var0
    const size_t OW_OFF  = 0;
    const size_t RAW_OFF = (size_t)BB * SH;                       // 2048
    const size_t MU_OFF  = RAW_OFF + (size_t)BB * SH * VV;        // +65,536,000
    const size_t LV_OFF  = MU_OFF + (size_t)BB * HH;

    // workspace layout (floats)
    float* ws = (float*)d_ws;
    float* emb    = ws;                                   // 2048*512 (prem, then dec reuse)
    float* big    = emb + (size_t)BB * SP * EE;           // 2048*4096 (x_pre, then pre1)
    float* h      = big + (size_t)BB * SP * H4;           // 32768
    float* c      = h + (size_t)BB * HH;                  // 32768
    float* pre    = c + (size_t)BB * HH;                  // 32*4096
    float* z0     = pre + (size_t)BB * H4;                // 32768
    float* zW     = z0 + (size_t)BB * HH;                 // 32*4096
    float* hdec   = zW + (size_t)BB * H4;                 // 2016*1024

    // ---- init: h=c=0 (contiguous), raw_outputs[:,0,:]=0 ----
    k_zero<<<(2 * BB * HH + 255) / 256, 256, 0, stream>>>(h, 2 * BB * HH);
    k_zero_row0<<<BB, 256, 0, stream>>>(out + RAW_OFF);

    // ---- encoder: embed premises, x_pre = emb @ Wih^T + bih + bhh ----
    k_gather<<<BB * SP, 128, 0, stream>>>(embed_W, premises, SP, SP, emb);
    k_gemm_xwT<true, true, false, false><<<dim3(H4 / BN, (BB * SP) / BM), 256, 0, stream>>>(
        emb, EE, enc_Wih, enc_bih, enc_bhh, nullptr, 0, big, H4, EE);

    // ---- encoder recurrence: 64 sequential steps ----
    for (int t = 0; t < SP; ++t) {
        k_gemm_xwT<false, false, true, false><<<dim3(H4 / BN, 1), 256, 0, stream>>>(
            h, HH, enc_Whh, nullptr, nullptr,
            big + (size_t)t * H4, SP * H4,     // Cadd = x_pre[:, t, :], row stride SP*4H
            pre, H4, HH);
        k_lstm_step<<<(BB * HH) / 256, 256, 0, stream>>>(pre, h, c);
    }

    // ---- latent: mu0, log_var0 straight into d_out; sample z0 ----
    k_gemm_xwT<true, false, false, false><<<dim3(HH / BN, 1), 256, 0, stream>>>(
        h, HH, mu_W, mu_b, nullptr, nullptr, 0, out + MU_OFF, HH, HH);
    k_gemm_xwT<true, false, false, false><<<dim3(HH / BN, 1), 256, 0, stream>>>(
        h, HH, lv_W, lv_b, nullptr, nullptr, 0, out + LV_OFF, HH, HH);
    k_sample_z<<<(BB * HH) / 256, 256, 0, stream>>>(out + MU_OFF, out + LV_OFF, z0);

    // ---- decoder (non-recurrent single gate step) ----
    k_gather<<<MDEC, 128, 0, stream>>>(embed_W, hypothesis, SH, SD, emb);
    k_gemm_xwT<true, true, false, false><<<dim3(H4 / BN, MDEC / BM), 256, 0, stream>>>(
        emb, EE, dec_Wih, dec_bih, dec_bhh, nullptr, 0, big, H4, EE);
    k_gemm_xwT<false, false, false, false><<<dim3(H4 / BN, 1), 256, 0, stream>>>(
        z0, HH, dec_Whh, nullptr, nullptr, nullptr, 0, zW, H4, HH);
    k_dec_gates<<<(MDEC * HH) / 256, 256, 0, stream>>>(big, zW, z0, hdec);

    // ---- logits: h_dec @ out_W^T + out_b -> raw_outputs[b, s+1, :] ----
    k_gemm_xwT<true, false, false, true><<<dim3(VV / BN, MDEC / BM), 256, 0, stream>>>(
        hdec, HH, out_W, out_b, nullptr, nullptr, 0, out + RAW_OFF, VV, HH);

    // ---- outputted_words ----
    k_argmax<<<MDEC, 256, 0, stream>>>(out + RAW_OFF, out + OW_OFF);
    k_ow0<<<1, 32, 0, stream>>>(hypothesis, out + OW_OFF);

    (void)in_sizes; (void)n_in; (void)out_size; (void)ws_size;
}